// SparseAutoencoder_38500086841370
// MI455X (gfx1250) — compile-verified
//
#include <hip/hip_runtime.h>

typedef __attribute__((ext_vector_type(2))) float v2f;
typedef __attribute__((ext_vector_type(8))) float v8f;

#define INPUT_DIM  512
#define HIDDEN_DIM 2048
#define TOP_K      50
#define NROWS      65536

// Block tile: 64 rows (M) x 32 cols, K chunked by 32. 256 threads = 8 waves,
// each wave owns one 16x16 C tile (4 M-subtiles x 2 col-subtiles).
#define BM 64
#define BN 32
#define BK 32
#define PAD 4   // +4 floats -> 144B row stride: 16B-aligned & bank-conflict-free

// Async copy of 16 bytes: global -> LDS, bypassing VGPRs (gfx1250).
__device__ __forceinline__ void async_b128(void* lds_dst, const void* gsrc) {
  const unsigned dst = (unsigned)(size_t)lds_dst;             // LDS offset
  const unsigned long long src = (unsigned long long)(size_t)gsrc;
  asm volatile("global_load_async_to_lds_b128 %0, %1, off"
               :: "v"(dst), "v"(src) : "memory");
}
__device__ __forceinline__ void wait_async0() {
  asm volatile("s_wait_asynccnt 0x0" ::: "memory");
}

// ---------------------------------------------------------------------------
// Encode: z = (x - b_dec - pre_bias) @ W_enc^T + b_enc       (N x HIDDEN_DIM)
// ---------------------------------------------------------------------------
__global__ __launch_bounds__(256) void sae_encode(
    const float* __restrict__ x, const float* __restrict__ W_enc,
    const float* __restrict__ b_enc, const float* __restrict__ b_dec,
    const float* __restrict__ pre_bias, float* __restrict__ z)
{
  __shared__ __align__(16) float ldsA[2][BM][BK + PAD];   // [row][k]
  __shared__ __align__(16) float ldsB[2][BN][BK + PAD];   // [col][k]

  const int h0   = blockIdx.x * BN;
  const int n0   = blockIdx.y * BM;
  const int tid  = threadIdx.x;
  const int lane = tid & 31;
  const int wave = tid >> 5;
  const int wm   = wave & 3;    // M subtile 0..3
  const int wn   = wave >> 2;   // col subtile 0..1
  const int colC = lane & 15;
  const int half = lane >> 4;   // 0 or 1

  v8f c;
  {
    const float bias = b_enc[h0 + wn * 16 + colC];
    #pragma unroll
    for (int i = 0; i < 8; ++i) c[i] = bias;
  }

  // --- tile helpers -------------------------------------------------------
  const int bcol = tid >> 3;            // 0..31
  const int bk4  = (tid & 7) << 2;      // 0,4,...,28

  auto issueB = [&](int kc, int buf) {  // one async B128 per thread
    async_b128(&ldsB[buf][bcol][bk4],
               &W_enc[(size_t)(h0 + bcol) * INPUT_DIM + kc + bk4]);
  };
  auto loadA = [&](int kc, float (&r)[8]) {
    #pragma unroll
    for (int i = 0; i < 8; ++i) {
      const int idx = tid + i * 256;
      const int rr = idx >> 5, k = idx & 31;
      r[i] = x[(size_t)(n0 + rr) * INPUT_DIM + kc + k]
             - b_dec[kc + k] - pre_bias[kc + k];
    }
  };
  auto storeA = [&](int buf, const float (&r)[8]) {
    #pragma unroll
    for (int i = 0; i < 8; ++i) {
      const int idx = tid + i * 256;
      ldsA[buf][idx >> 5][idx & 31] = r[i];
    }
  };
  auto compute = [&](int buf) {
    #pragma unroll
    for (int kk = 0; kk < BK; kk += 4) {
      v2f a, b;
      const int ra = wm * 16 + colC;
      const int ka = kk + half * 2;
      a.x = ldsA[buf][ra][ka];
      a.y = ldsA[buf][ra][ka + 1];
      const int cbn = wn * 16 + colC;
      b.x = ldsB[buf][cbn][ka];
      b.y = ldsB[buf][cbn][ka + 1];
      c = __builtin_amdgcn_wmma_f32_16x16x4_f32(false, a, false, b,
                                                (short)0, c, false, false);
    }
  };

  // --- prologue: tile 0 ---------------------------------------------------
  issueB(0, 0);
  { float r[8]; loadA(0, r); storeA(0, r); }
  wait_async0();
  __syncthreads();

  // --- pipelined main loop ------------------------------------------------
  const int T = INPUT_DIM / BK;                 // 16
  for (int t = 0; t < T; ++t) {
    const int cur = t & 1, nxt = cur ^ 1;
    if (t + 1 < T) {
      issueB((t + 1) * BK, nxt);                // async copy overlaps WMMA
      float r[8]; loadA((t + 1) * BK, r); storeA(nxt, r);
    }
    compute(cur);
    wait_async0();
    __syncthreads();
  }

  #pragma unroll
  for (int i = 0; i < 8; ++i) {
    const int r = wm * 16 + half * 8 + i;
    z[(size_t)(n0 + r) * HIDDEN_DIM + h0 + wn * 16 + colC] = c[i];
  }
}

// ---------------------------------------------------------------------------
// Exact 50th-largest per row via 4-pass radix select on monotone u32 keys,
// then in-place masking z = (z >= thr) ? z : 0. One 256-thread block per row.
// ---------------------------------------------------------------------------
__global__ __launch_bounds__(256) void sae_topk_sparsify(float* __restrict__ z)
{
  __shared__ unsigned hist[256];
  __shared__ unsigned s_prefix, s_remaining;

  float* zr = z + (size_t)blockIdx.x * HIDDEN_DIM;
  const int tid = threadIdx.x;

  unsigned keys[HIDDEN_DIM / 256];
  #pragma unroll
  for (int i = 0; i < HIDDEN_DIM / 256; ++i) {
    const unsigned b = __float_as_uint(zr[tid + i * 256]);
    keys[i] = (b & 0x80000000u) ? ~b : (b | 0x80000000u);
  }
  if (tid == 0) { s_prefix = 0u; s_remaining = TOP_K; }
  __syncthreads();

  for (int pass = 3; pass >= 0; --pass) {
    hist[tid] = 0u;
    __syncthreads();
    const unsigned prefix = s_prefix;
    const unsigned shift  = (unsigned)pass * 8u;
    const unsigned hiMask = (pass == 3) ? 0u : (0xFFFFFFFFu << (shift + 8u));
    #pragma unroll
    for (int i = 0; i < HIDDEN_DIM / 256; ++i)
      if ((keys[i] & hiMask) == (prefix & hiMask))
        atomicAdd(&hist[(keys[i] >> shift) & 0xFFu], 1u);
    __syncthreads();
    if (tid == 0) {
      const unsigned rem = s_remaining;
      unsigned cum = 0;
      int b = 255;
      for (; b > 0; --b) {
        const unsigned h = hist[b];
        if (cum + h >= rem) break;
        cum += h;
      }
      s_prefix    = prefix | ((unsigned)b << shift);
      s_remaining = rem - cum;
    }
    __syncthreads();
  }

  const unsigned kkey = s_prefix;
  const unsigned bits = (kkey & 0x80000000u) ? (kkey & 0x7FFFFFFFu) : ~kkey;
  const float thr = __uint_as_float(bits);
  #pragma unroll
  for (int i = 0; i < HIDDEN_DIM / 256; ++i) {
    const int idx = tid + i * 256;
    const float v = zr[idx];
    zr[idx] = (v >= thr) ? v : 0.0f;
  }
}

// ---------------------------------------------------------------------------
// Decode: recon = z_sparse @ W_dec^T + b_dec                  (N x INPUT_DIM)
// recon[n,d] = sum_h z[n,h] * W_dec[d,h].  Both A and B tiles copied async.
// ---------------------------------------------------------------------------
__global__ __launch_bounds__(256) void sae_decode(
    const float* __restrict__ zs, const float* __restrict__ W_dec,
    const float* __restrict__ b_dec, float* __restrict__ recon)
{
  __shared__ __align__(16) float ldsA[2][BM][BK + PAD];   // [row][k]
  __shared__ __align__(16) float ldsB[2][BN][BK + PAD];   // [col][k]

  const int d0   = blockIdx.x * BN;
  const int n0   = blockIdx.y * BM;
  const int tid  = threadIdx.x;
  const int lane = tid & 31;
  const int wave = tid >> 5;
  const int wm   = wave & 3;
  const int wn   = wave >> 2;
  const int colC = lane & 15;
  const int half = lane >> 4;

  v8f c;
  {
    const float bias = b_dec[d0 + wn * 16 + colC];
    #pragma unroll
    for (int i = 0; i < 8; ++i) c[i] = bias;
  }

  const int bcol = tid >> 3;
  const int bk4  = (tid & 7) << 2;

  auto issueTile = [&](int kc, int buf) {     // 3 async B128 per thread
    #pragma unroll
    for (int i = 0; i < 2; ++i) {             // A: 64x32 floats
      const int g = tid + i * 256;
      const int r = g >> 3, k4 = (g & 7) << 2;
      async_b128(&ldsA[buf][r][k4],
                 &zs[(size_t)(n0 + r) * HIDDEN_DIM + kc + k4]);
    }
    async_b128(&ldsB[buf][bcol][bk4],         // B: 32x32 floats
               &W_dec[(size_t)(d0 + bcol) * HIDDEN_DIM + kc + bk4]);
  };
  auto compute = [&](int buf) {
    #pragma unroll
    for (int kk = 0; kk < BK; kk += 4) {
      v2f a, b;
      const int ra = wm * 16 + colC;
      const int ka = kk + half * 2;
      a.x = ldsA[buf][ra][ka];
      a.y = ldsA[buf][ra][ka + 1];
      const int cbn = wn * 16 + colC;
      b.x = ldsB[buf][cbn][ka];
      b.y = ldsB[buf][cbn][ka + 1];
      c = __builtin_amdgcn_wmma_f32_16x16x4_f32(false, a, false, b,
                                                (short)0, c, false, false);
    }
  };

  issueTile(0, 0);
  wait_async0();
  __syncthreads();

  const int T = HIDDEN_DIM / BK;              // 64
  for (int t = 0; t < T; ++t) {
    const int cur = t & 1, nxt = cur ^ 1;
    if (t + 1 < T) issueTile((t + 1) * BK, nxt);
    compute(cur);
    wait_async0();
    __syncthreads();
  }

  #pragma unroll
  for (int i = 0; i < 8; ++i) {
    const int r = wm * 16 + half * 8 + i;
    recon[(size_t)(n0 + r) * INPUT_DIM + d0 + wn * 16 + colC] = c[i];
  }
}

// ---------------------------------------------------------------------------
extern "C" void kernel_launch(void* const* d_in, const int* in_sizes, int n_in,
                              void* d_out, int out_size, void* d_ws, size_t ws_size,
                              hipStream_t stream) {
  (void)in_sizes; (void)n_in; (void)out_size; (void)d_ws; (void)ws_size;
  const float* x        = (const float*)d_in[0];
  const float* W_enc    = (const float*)d_in[1];
  const float* b_enc    = (const float*)d_in[2];
  const float* W_dec    = (const float*)d_in[3];
  const float* b_dec    = (const float*)d_in[4];
  const float* pre_bias = (const float*)d_in[5];

  float* recon = (float*)d_out;                              // N * 512
  float* z     = (float*)d_out + (size_t)NROWS * INPUT_DIM;  // N * 2048 (z_sparse)

  dim3 gridE(HIDDEN_DIM / BN, NROWS / BM);
  sae_encode<<<gridE, 256, 0, stream>>>(x, W_enc, b_enc, b_dec, pre_bias, z);

  sae_topk_sparsify<<<NROWS, 256, 0, stream>>>(z);

  dim3 gridD(INPUT_DIM / BN, NROWS / BM);
  sae_decode<<<gridD, 256, 0, stream>>>(z, W_dec, b_dec, recon);
}